// GCN_31464930410621
// MI455X (gfx1250) — compile-verified
//
#include <hip/hip_runtime.h>
#include <math.h>

// ---------------------------------------------------------------------------
// GCN 2-layer forward for gfx1250 (MI455X).
//   layer: h = A @ W ; msg = h[src]*dinv[src]*dinv[dst]; agg = scatter_add(msg)
//          out = relu(agg + selfloop + b)
//   final: softmax over 16 classes.
// GEMMs use V_WMMA_F32_16X16X4_F32 (exact fp32 matrix path).
// Edge phase is L2-resident (h1=19.2MB, h2=6.4MB << 192MB L2) and bound by
// f32 atomic-add throughput at L2, not HBM. Edge messages move as float4
// chunks: one b128 gather + 4 contiguous f32 atomics per thread.
// ---------------------------------------------------------------------------

typedef __attribute__((ext_vector_type(2))) float v2f;
typedef __attribute__((ext_vector_type(4))) float v4f;
typedef __attribute__((ext_vector_type(8))) float v8f;

__device__ __forceinline__ void atomic_add_f32(float* p, float v) {
    unsafeAtomicAdd(p, v);   // native global_atomic_add_f32 on AMD
}

// ------------------------------ utility ------------------------------------

__global__ void fill_kernel(float* __restrict__ p, float v, long long n) {
    long long i = (long long)blockIdx.x * blockDim.x + threadIdx.x;
    long long stride = (long long)gridDim.x * blockDim.x;
    for (; i < n; i += stride) p[i] = v;
}

__global__ void degree_kernel(const int* __restrict__ dst,
                              float* __restrict__ deg, long long E) {
    long long i = (long long)blockIdx.x * blockDim.x + threadIdx.x;
    long long stride = (long long)gridDim.x * blockDim.x;
    for (; i < E; i += stride) atomic_add_f32(&deg[dst[i]], 1.0f);
}

__global__ void rsqrt_kernel(float* __restrict__ d, int n) {
    int i = blockIdx.x * blockDim.x + threadIdx.x;
    if (i < n) d[i] = rsqrtf(d[i]);   // deg >= 1 always (self-loop baked in)
}

// ------------------------------ WMMA GEMM ----------------------------------
// C[nrows x (NT*16)] = A[nrows x K] @ B[K x (NT*16)], all fp32 row-major.
// One wave computes a 16-row stripe across all NT column tiles.
// A-frag (16x4 f32): lanes 0-15 -> K = k0+0/+1 ; lanes 16-31 -> K = k0+2/+3.
// B-frag (4x16 f32): same K split across lane halves, N = lane&15.
// D (16x16 f32, 8 VGPRs): VGPR r -> row m0 + r + 8*(lane>=16), col = lane&15.
// Out-of-range A rows are CLAMPED (not predicated): a bad A row M only
// pollutes D row M, which is never stored -> no EXEC-divergent loads, and the
// two contiguous A elements load as one global_load_b64.
template<int K, int NT>
__global__ void gemm_wmma_kernel(const float* __restrict__ A,
                                 const float* __restrict__ B,
                                 float* __restrict__ C, int nrows) {
    const int NCOL   = NT * 16;
    const int wave   = threadIdx.x >> 5;
    const int lane   = threadIdx.x & 31;
    const int m0     = (blockIdx.x * (blockDim.x >> 5) + wave) * 16;
    if (m0 >= nrows) return;                       // wave-uniform exit

    const int laneRow = lane & 15;                 // M (A) / N (B,D) within tile
    const int kOff    = (lane >> 4) * 2;           // 0 for lanes 0-15, 2 for 16-31
    int row = m0 + laneRow;
    if (row >= nrows) row = nrows - 1;             // clamp: garbage row is discarded
    const float* __restrict__ Arow = A + (long long)row * K + kOff;

    v8f acc[NT];
#pragma unroll
    for (int t = 0; t < NT; ++t)
#pragma unroll
        for (int r = 0; r < 8; ++r) acc[t][r] = 0.0f;

#pragma unroll 4
    for (int k0 = 0; k0 < K; k0 += 4) {
        v2f a = *(const v2f*)(Arow + k0);          // 8B-aligned: one b64 load
#pragma unroll
        for (int t = 0; t < NT; ++t) {
            v2f b;
            b.x = B[(k0 + kOff)     * NCOL + t * 16 + laneRow];
            b.y = B[(k0 + kOff + 1) * NCOL + t * 16 + laneRow];
            acc[t] = __builtin_amdgcn_wmma_f32_16x16x4_f32(
                false, a, false, b, (short)0, acc[t], false, false);
        }
    }

    const int rowHalf = (lane >> 4) * 8;
#pragma unroll
    for (int t = 0; t < NT; ++t)
#pragma unroll
        for (int r = 0; r < 8; ++r) {
            int rr = m0 + rowHalf + r;
            if (rr < nrows)
                C[(long long)rr * NCOL + t * 16 + laneRow] = acc[t][r];
        }
}

// ------------------------------ edge scatter -------------------------------
// One float4 chunk of one edge message per thread. F multiple of 4, so every
// chunk is 16B-aligned: one global_load_b128 gather + 4 f32 atomics (L2).
// C4 = F/4 chunks per edge; chunk-major within edge -> contiguous atomics.
template<int F>
__global__ void scatter_kernel(const float* __restrict__ h,
                               const int* __restrict__ src,
                               const int* __restrict__ dst,
                               const float* __restrict__ dinv,
                               float* __restrict__ out, long long total4) {
    constexpr int C4 = F / 4;
    long long i = (long long)blockIdx.x * blockDim.x + threadIdx.x;
    long long stride = (long long)gridDim.x * blockDim.x;
    for (; i < total4; i += stride) {
        long long e = i / C4;
        int f = (int)(i - e * C4) * 4;
        int s = src[e];
        int d = dst[e];
        float nrm = dinv[s] * dinv[d];
        v4f m = *(const v4f*)(h + (long long)s * F + f);
        float* o = out + (long long)d * F + f;
        atomic_add_f32(o + 0, m.x * nrm);
        atomic_add_f32(o + 1, m.y * nrm);
        atomic_add_f32(o + 2, m.z * nrm);
        atomic_add_f32(o + 3, m.w * nrm);
    }
}

// ----------------------- self-loop + bias + relu ---------------------------
template<int F>
__global__ void bias_relu_self_kernel(float* __restrict__ agg,
                                      const float* __restrict__ h,
                                      const float* __restrict__ dinv,
                                      const float* __restrict__ bias,
                                      int nrows) {
    long long total = (long long)nrows * F;
    long long i = (long long)blockIdx.x * blockDim.x + threadIdx.x;
    long long stride = (long long)gridDim.x * blockDim.x;
    for (; i < total; i += stride) {
        long long n = i / F;
        int f = (int)(i - n * F);
        float di = dinv[n];
        float v = agg[i] + h[i] * di * di + bias[f];
        agg[i] = v > 0.0f ? v : 0.0f;
    }
}

// --------------- layer-2 epilogue: self-loop+bias+relu+softmax -------------
__global__ void final_kernel(float* __restrict__ out,
                             const float* __restrict__ h2,
                             const float* __restrict__ dinv,
                             const float* __restrict__ b2, int nrows) {
    int n = blockIdx.x * blockDim.x + threadIdx.x;
    if (n >= nrows) return;
    float dd = dinv[n] * dinv[n];
    long long base = (long long)n * 16;
    float v[16];
    float mx = -INFINITY;
#pragma unroll
    for (int f = 0; f < 16; ++f) {
        float t = out[base + f] + h2[base + f] * dd + b2[f];
        t = t > 0.0f ? t : 0.0f;
        v[f] = t;
        mx = fmaxf(mx, t);
    }
    float s = 0.0f;
#pragma unroll
    for (int f = 0; f < 16; ++f) { v[f] = expf(v[f] - mx); s += v[f]; }
    float inv = 1.0f / s;
#pragma unroll
    for (int f = 0; f < 16; ++f) out[base + f] = v[f] * inv;
}

// ------------------------------ launcher -----------------------------------

extern "C" void kernel_launch(void* const* d_in, const int* in_sizes, int n_in,
                              void* d_out, int out_size, void* d_ws, size_t ws_size,
                              hipStream_t stream) {
    const float* x  = (const float*)d_in[0];
    const int*   ei = (const int*)d_in[1];     // [2, E] int32 (JAX default x32)
    const float* W1 = (const float*)d_in[2];   // [128, 48]
    const float* b1 = (const float*)d_in[3];   // [48]
    const float* W2 = (const float*)d_in[4];   // [48, 16]
    const float* b2 = (const float*)d_in[5];   // [16]
    float* out = (float*)d_out;                // [N, 16]

    const int       nNodes = in_sizes[0] / 128;
    const long long E      = in_sizes[1] / 2;
    const int* src = ei;
    const int* dst = ei + E;

    // workspace carve-up (hipMalloc'd -> base is aligned)
    float* ws   = (float*)d_ws;
    long long nPad = ((long long)nNodes + 255) & ~255LL;
    float* dinv = ws;                          // nNodes (init 1.0 = self loop)
    float* h1   = dinv + nPad;                 // nNodes*48 ; reused as h2 later
    float* agg1 = h1 + (long long)nNodes * 48; // nNodes*48
    float* h2   = h1;                          // safe: h1 dead after epilogue-1

    const int B = 256;
    auto blocks = [](long long n, int b) { return (unsigned)((n + b - 1) / b); };

    // 1) init: deg=1 (self loop), agg1=0, out=0
    fill_kernel<<<blocks(nNodes, B), B, 0, stream>>>(dinv, 1.0f, nNodes);
    fill_kernel<<<blocks((long long)nNodes * 48, B), B, 0, stream>>>(agg1, 0.0f, (long long)nNodes * 48);
    fill_kernel<<<blocks((long long)nNodes * 16, B), B, 0, stream>>>(out, 0.0f, (long long)nNodes * 16);

    // 2) degree + 3) dinv = rsqrt(deg)
    degree_kernel<<<blocks(E, B), B, 0, stream>>>(dst, dinv, E);
    rsqrt_kernel<<<blocks(nNodes, B), B, 0, stream>>>(dinv, nNodes);

    // 4) h1 = x @ W1   (N x 128 @ 128 x 48) — WMMA f32, 4 waves * 16 rows / block
    gemm_wmma_kernel<128, 3><<<blocks(nNodes, 64), 128, 0, stream>>>(x, W1, h1, nNodes);

    // 5) edge aggregate layer 1 (F=48): E * 12 float4-chunks
    scatter_kernel<48><<<blocks(E * 12, B), B, 0, stream>>>(h1, src, dst, dinv, agg1, E * 12);

    // 6) agg1 = relu(agg1 + h1*dinv^2 + b1)
    bias_relu_self_kernel<48><<<blocks((long long)nNodes * 48, B), B, 0, stream>>>(agg1, h1, dinv, b1, nNodes);

    // 7) h2 = agg1 @ W2  (N x 48 @ 48 x 16)
    gemm_wmma_kernel<48, 1><<<blocks(nNodes, 64), 128, 0, stream>>>(agg1, W2, h2, nNodes);

    // 8) edge aggregate layer 2 (F=16) directly into d_out: E * 4 chunks
    scatter_kernel<16><<<blocks(E * 4, B), B, 0, stream>>>(h2, src, dst, dinv, out, E * 4);

    // 9) out = softmax(relu(out + h2*dinv^2 + b2))
    final_kernel<<<blocks(nNodes, B), B, 0, stream>>>(out, h2, dinv, b2, nNodes);
}